// Resize_27333171872126
// MI455X (gfx1250) — compile-verified
//
#include <hip/hip_runtime.h>

typedef float v2f __attribute__((ext_vector_type(2)));
typedef float v8f __attribute__((ext_vector_type(8)));

#define IN_D   96
#define OUT_D  192
#define NCH    2
#define H_STRIDE   (IN_D * NCH)            // 192 floats per (d,h) row
#define D_STRIDE   (IN_D * H_STRIDE)       // 18432
#define B_STRIDE   ((size_t)IN_D * D_STRIDE)
#define C_PITCH    100                      // staged per-channel row (97 used + pad)
#define J_PITCH    204                      // per-ho-row pitch (bank-conflict free: 204%64=12)
#define WAVE_LDS   (16 * J_PITCH)           // floats per wave
#define WAVES_PER_BLOCK 4

__device__ __forceinline__ float wmat(int ki, int n) {
    // weight of staged input ki for output column n within a 16-wide output tile
    // (inputs indexed 0..11 across the 3 K-chunks; only 0..8 ever nonzero)
    if (n & 1) return (ki == (n >> 1) || ki == (n >> 1) + 1) ? 0.5f : 0.0f;
    return (ki == (n >> 1)) ? 1.0f : 0.0f;
}

__global__ __launch_bounds__(128)
void resize_zoom2_wmma(const float* __restrict__ in, float* __restrict__ out) {
    __shared__ float lds[WAVES_PER_BLOCK * WAVE_LDS];

    const int lane = threadIdx.x & 31;
    const int wave = threadIdx.x >> 5;
    float* st = lds + wave * WAVE_LDS;

    // one (b, d_out, ho-block) unit per wave: 4 * 192 * 12 = 9216 units
    const int unit = blockIdx.x * WAVES_PER_BLOCK + wave;
    const int hb  = unit % 12;
    const int t0  = unit / 12;
    const int dz  = t0 % OUT_D;     // output d
    const int b   = t0 / OUT_D;

    const int  d0   = dz >> 1;
    const bool dodd = (dz & 1) != 0;
    const int  d1   = (d0 + 1 < IN_D) ? d0 + 1 : IN_D - 1;

    const float* inb   = in + (size_t)b * B_STRIDE;
    const float* based0 = inb + (size_t)d0 * D_STRIDE;
    const float* based1 = inb + (size_t)d1 * D_STRIDE;

    // ---- stage 16 d/h-blended rows into LDS: layout [j][c:C_PITCH][w] ----
    for (int j = 0; j < 16; ++j) {
        const int  ho   = hb * 16 + j;
        const int  h0   = ho >> 1;
        const bool hodd = (ho & 1) != 0;
        const int  h1   = (h0 + 1 < IN_D) ? h0 + 1 : IN_D - 1;
        const float* p00 = based0 + h0 * H_STRIDE;
        const float* p10 = based1 + h0 * H_STRIDE;
        const float* p01 = based0 + h1 * H_STRIDE;
        const float* p11 = based1 + h1 * H_STRIDE;
        for (int r = lane; r < 2 * C_PITCH; r += 32) {
            const int w = r >> 1;
            const int c = r & 1;
            float v;
            if (w >= IN_D + 1) {
                v = 0.0f;                      // pad lanes (zero so 0-weight * v stays 0)
            } else {
                const int we  = (w < IN_D) ? w : IN_D - 1;   // w==96 -> clamp to 95
                const int off = we * NCH + c;
                v = p00[off];
                if (dodd) v = 0.5f * (v + p10[off]);
                if (hodd) {
                    float u = p01[off];
                    if (dodd) u = 0.5f * (u + p11[off]);
                    v = 0.5f * (v + u);
                }
            }
            st[j * J_PITCH + c * C_PITCH + w] = v;
        }
    }
    __syncthreads();

    // ---- build constant weight B-matrices (4x16, K x N), 3 K-chunks ----
    // B striping: lanes 0-15 hold K=0 (v.x), K=1 (v.y); lanes 16-31 hold K=2,K=3
    const int n  = lane & 15;   // output column (wo within tile)
    const int hi = lane >> 4;
    v2f Bw[3];
#pragma unroll
    for (int t = 0; t < 3; ++t) {
        const int kx = 4 * t + 2 * hi;
        Bw[t].x = wmat(kx,     n);
        Bw[t].y = wmat(kx + 1, n);
    }

    // A striping (16x4, M x K): lane m = lane&15, v.x: K = 2*hi, v.y: K = 2*hi+1
    const int m = lane & 15;
    const v2f* stv = (const v2f*)st;
    const int abase = m * (J_PITCH / 2);     // float2 units

    // output base: float2 (both channels) at ((b*192+dz)*192 + hb*16 + ho)*192 + wo
    float2* op = (float2*)out + ((size_t)(b * OUT_D + dz) * OUT_D + hb * 16) * OUT_D;

    for (int mt = 0; mt < 12; ++mt) {
        const int wb = 8 * mt;               // input-w base of this tile
        v8f acc0 = {0.f,0.f,0.f,0.f,0.f,0.f,0.f,0.f};
        v8f acc1 = {0.f,0.f,0.f,0.f,0.f,0.f,0.f,0.f};
#pragma unroll
        for (int t = 0; t < 3; ++t) {
            const int w0 = wb + 4 * t + 2 * hi;          // even -> aligned float2
            const v2f a0 = stv[abase + (w0 >> 1)];                 // channel 0
            const v2f a1 = stv[abase + (C_PITCH >> 1) + (w0 >> 1)];// channel 1
            acc0 = __builtin_amdgcn_wmma_f32_16x16x4_f32(
                false, a0, false, Bw[t], (short)0, acc0, false, false);
            acc1 = __builtin_amdgcn_wmma_f32_16x16x4_f32(
                false, a1, false, Bw[t], (short)0, acc1, false, false);
        }
        // D layout: lane holds column n; VGPR r holds row M = r + 8*hi (= ho_local)
#pragma unroll
        for (int r = 0; r < 8; ++r) {
            const int hoL = r + 8 * hi;
            float2 v; v.x = acc0[r]; v.y = acc1[r];
            op[(size_t)hoL * OUT_D + 2 * wb + n] = v;    // wo = 16*mt + n
        }
    }
}

extern "C" void kernel_launch(void* const* d_in, const int* in_sizes, int n_in,
                              void* d_out, int out_size, void* d_ws, size_t ws_size,
                              hipStream_t stream) {
    (void)in_sizes; (void)n_in; (void)d_ws; (void)ws_size; (void)out_size;
    const float* in = (const float*)d_in[0];
    float* out = (float*)d_out;
    // 4 batches * 192 d_out * 12 ho-blocks = 9216 wave-units, 4 waves/block
    dim3 grid(9216 / WAVES_PER_BLOCK), block(128);
    hipLaunchKernelGGL(resize_zoom2_wmma, grid, block, 0, stream, in, out);
}